// GW_Conv1D_89687507076122
// MI455X (gfx1250) — compile-verified
//
#include <hip/hip_runtime.h>
#include <hip/hip_bf16.h>

typedef __attribute__((ext_vector_type(2))) float v2f;
typedef __attribute__((ext_vector_type(8))) float v8f;

#define WSZ 256        // window size
#define NC 64          // channels
#define KDIM 192       // NC * 3 (im2col K, k-major: j = k*64 + ci)
#define NCOLS 258      // WSZ + 2 halo time columns
#define CSTR 68        // xst column stride: 68 % 64 == 4 -> conflict-free b64
#define WS_STRIDE 196  // KDIM padded: 196 % 64 == 4 (conflict-free b64 A fetch)
#define YS_STRIDE 258  // ys row stride: 8*258 % 64 == 16 (conflict-free store)

// ---------------------------------------------------------------------------
// Kernel 1: g = tanh(sig . est_w)*0.5 ; integrate +/-g by scaling & squaring.
// One 256-thread workgroup per (b, f) window.
// ---------------------------------------------------------------------------
__global__ __launch_bounds__(256) void stgw_flow_kernel(
    const float* __restrict__ sig, const float* __restrict__ est_w,
    float* __restrict__ d_fwd, float* __restrict__ d_inv, int F, int T) {
  __shared__ float ds[WSZ];
  const int win = blockIdx.x;
  const int b = win / F, f = win % F;
  const int w = threadIdx.x;
  const int t = f * WSZ + w;

  const float* sp = sig + (size_t)b * NC * T + t;
  float acc = 0.f;
#pragma unroll 8
  for (int c = 0; c < NC; ++c) acc = fmaf(sp[(size_t)c * T], est_w[c], acc);
  const float g = tanhf(acc) * 0.5f;

  for (int s = 0; s < 2; ++s) {
    float d = (s ? -g : g) * 0.0625f;  // g / 2^N, N = 4
    for (int it = 0; it < 4; ++it) {
      ds[w] = d;
      __syncthreads();
      float pos = fminf(fmaxf((float)w + d, 0.f), (float)(WSZ - 1));
      int i0 = (int)pos;
      int i1 = min(i0 + 1, WSZ - 1);
      float fr = pos - (float)i0;
      float dn = d + ds[i0] * (1.f - fr) + ds[i1] * fr;
      __syncthreads();
      d = dn;
    }
    (s ? d_inv : d_fwd)[(size_t)win * WSZ + w] = d;
  }
}

// ---------------------------------------------------------------------------
// Kernel 2: forward warp into time-major LDS tile, 64x256 conv-as-GEMM via
// V_WMMA_F32_16X16X4_F32 (K = 192 = 48 steps of 4), inverse warp, store.
// One 256-thread (8 wave32) workgroup per window.
// xst is time-major (xst[col][ci]) so every WMMA B operand pair (ci, ci+1)
// is one contiguous ds_load_b64 at a compile-time immediate offset.
// ---------------------------------------------------------------------------
__global__ __launch_bounds__(256) void stgw_warp_conv_kernel(
    const float* __restrict__ sig, const float* __restrict__ convw,
    const float* __restrict__ convb, const float* __restrict__ d_fwd,
    const float* __restrict__ d_inv, float* __restrict__ out, int F, int T) {
  extern __shared__ float lds[];
  float* xst = lds;                      // [NCOLS][CSTR]; col0 = time t0-1
  float* wsm = xst + NCOLS * CSTR;       // [NC][WS_STRIDE] k-major weights
  float* ys = wsm + NC * WS_STRIDE;      // [NC][YS_STRIDE] conv output

  const int win = blockIdx.x;
  const int b = win / F, f = win % F;
  const int tid = threadIdx.x;
  const int t0 = f * WSZ;
  const float* sb = sig + (size_t)b * NC * T;

  // stage weights permuted to k-major: wsm[co][k*64+ci] = conv_w[co][ci][k]
  for (int i = tid; i < NC * KDIM; i += 256) {
    const int co = i / KDIM;
    const int r = i - co * KDIM;
    const int k = r >> 6;
    const int ci = r & 63;
    wsm[co * WS_STRIDE + r] = convw[(co * NC + ci) * 3 + k];
  }

  // ---- forward warp: thread tid owns time column t0+tid (xst col 1+tid) ---
  {
    float dv = d_fwd[(size_t)win * WSZ + tid];
    float pos = fminf(fmaxf((float)tid + dv, 0.f), (float)(WSZ - 1));
    int i0 = (int)pos;
    int i1 = min(i0 + 1, WSZ - 1);
    float fr = pos - (float)i0;
    float* xr = xst + (1 + tid) * CSTR;  // contiguous 64-float row
    for (int c = 0; c < NC; ++c) {
      const float* cb = sb + (size_t)c * T + t0;
      xr[c] = cb[i0] * (1.f - fr) + cb[i1] * fr;
    }
  }
  // ---- halo columns (SAME conv crosses window edges); warped with the
  //      neighbor window's displacement, zero at sequence boundaries ----
  if (tid < NC) {  // left halo (time t0-1), channel = tid
    float v = 0.f;
    if (f > 0) {
      float dv = d_fwd[(size_t)(win - 1) * WSZ + (WSZ - 1)];
      float pos = fminf(fmaxf((float)(WSZ - 1) + dv, 0.f), (float)(WSZ - 1));
      int i0 = (int)pos;
      int i1 = min(i0 + 1, WSZ - 1);
      float fr = pos - (float)i0;
      const float* cb = sb + (size_t)tid * T + (t0 - WSZ);
      v = cb[i0] * (1.f - fr) + cb[i1] * fr;
    }
    xst[tid] = v;  // col 0
  } else if (tid < 2 * NC) {  // right halo (time t0+WSZ), channel = tid - NC
    const int c = tid - NC;
    float v = 0.f;
    if (f + 1 < F) {
      float dv = d_fwd[(size_t)(win + 1) * WSZ];
      float pos = fminf(fmaxf(dv, 0.f), (float)(WSZ - 1));
      int i0 = (int)pos;
      int i1 = min(i0 + 1, WSZ - 1);
      float fr = pos - (float)i0;
      const float* cb = sb + (size_t)c * T + (t0 + WSZ);
      v = cb[i0] * (1.f - fr) + cb[i1] * fr;
    }
    xst[(1 + WSZ) * CSTR + c] = v;  // col 257
  }
  __syncthreads();

  // ---- GEMM: D(64x256) = W(64x192) x im2col(192x256) + bias -------------
  // A 16x4 f32: lanes 0-15 row M, VGPR0/1 = K0/K1; lanes 16-31 = K2/K3.
  // B 4x16 f32 mirrored; C/D: VGPR r -> M = r (+8 for lanes 16-31).
  // K index j = 4*kk + 2*khalf (+1 for .y): ci = j & 63, ktap = j >> 6.
  // No carry into bit 6 within a pair => ktap = kk >> 4 (const per step),
  // ci pair (ci0, ci0+1) contiguous in xst => one immediate-offset b64.
  const int wv = tid >> 5;
  const int lane = tid & 31;
  const int l15 = lane & 15;
  const int khalf = lane >> 4;  // 0: K pair {j0,j0+1}, 1: pair {j0+2,j0+3}
  for (int i = 0; i < 8; ++i) {
    const int tile = wv * 8 + i;
    const int mbase = (tile & 3) * 16;
    const int nbase = (tile >> 2) * 16;
    v8f acc;
#pragma unroll
    for (int r = 0; r < 8; ++r) acc[r] = convb[mbase + r + khalf * 8];
    const float* wsA = wsm + (mbase + l15) * WS_STRIDE + 2 * khalf;
    // B base: time col (nbase + l15), channel 2*khalf (col0 of tile is +k)
    const float* xsB = xst + (nbase + l15) * CSTR + 2 * khalf;
#pragma unroll
    for (int kk = 0; kk < 48; ++kk) {
      const int boff = (kk >> 4) * CSTR + ((4 * kk) & 63);  // tap*col + ci
      v2f a = *(const v2f*)(wsA + 4 * kk);   // ds_load_b64, imm offset
      v2f bb = *(const v2f*)(xsB + boff);    // ds_load_b64, imm offset
      acc = __builtin_amdgcn_wmma_f32_16x16x4_f32(
          false, a, false, bb, (short)0, acc, false, false);
    }
    // store this tile's conv output (channel-major for the inverse gather)
#pragma unroll
    for (int r = 0; r < 8; ++r)
      ys[(mbase + r + 8 * khalf) * YS_STRIDE + nbase + l15] = acc[r];
  }
  __syncthreads();

  // ---- inverse warp (pure LDS gather), coalesced global store -----------
  {
    float dv = d_inv[(size_t)win * WSZ + tid];
    float pos = fminf(fmaxf((float)tid + dv, 0.f), (float)(WSZ - 1));
    int i0 = (int)pos;
    int i1 = min(i0 + 1, WSZ - 1);
    float fr = pos - (float)i0;
    float* ob = out + (size_t)b * NC * T + t0 + tid;
    for (int c = 0; c < NC; ++c)
      ob[(size_t)c * T] =
          ys[c * YS_STRIDE + i0] * (1.f - fr) + ys[c * YS_STRIDE + i1] * fr;
  }
}

// ---------------------------------------------------------------------------
extern "C" void kernel_launch(void* const* d_in, const int* in_sizes, int n_in,
                              void* d_out, int out_size, void* d_ws,
                              size_t ws_size, hipStream_t stream) {
  const float* sig = (const float*)d_in[0];    // (B, C, T) f32
  const float* estw = (const float*)d_in[1];   // (C,)
  const float* convw = (const float*)d_in[2];  // (C, C, 3)
  const float* convb = (const float*)d_in[3];  // (C,)
  float* out = (float*)d_out;                  // (B, C, T) f32

  const int C = NC;
  const int T = 32768;
  const int B = in_sizes[0] / (C * T);
  const int F = T / WSZ;  // T already a multiple of WSZ (pad == 0)
  const int nwin = B * F;

  float* dfwd = (float*)d_ws;               // (B*F, W)
  float* dinv = dfwd + (size_t)nwin * WSZ;  // (B*F, W)

  stgw_flow_kernel<<<nwin, 256, 0, stream>>>(sig, estw, dfwd, dinv, F, T);

  const size_t shmem =
      (size_t)(NCOLS * CSTR + NC * WS_STRIDE + NC * YS_STRIDE) *
      sizeof(float);
  (void)hipFuncSetAttribute((const void*)stgw_warp_conv_kernel,
                            hipFuncAttributeMaxDynamicSharedMemorySize,
                            (int)shmem);
  stgw_warp_conv_kernel<<<nwin, 256, shmem, stream>>>(sig, convw, convb, dfwd,
                                                      dinv, out, F, T);
}